// SynergyGCN_48155173322903
// MI455X (gfx1250) — compile-verified
//
#include <hip/hip_runtime.h>
#include <math.h>

#define DIN 64
#define DH  128
#define EPSBN 1e-5f

typedef __attribute__((ext_vector_type(2))) float v2f;
typedef __attribute__((ext_vector_type(8))) float v8f;

// ---------------------------------------------------------------------------
// GEMM: Y[N x DH] = X[N x K] @ W[K x DH], fp32 via V_WMMA_F32_16X16X4_F32.
// Block = 256 threads = 8 waves. Block tile: 64 rows x 128 cols.
// Wave w -> column tile w (16 cols); 4 row tiles of 16 per wave.
// B fragments (full K depth) live in registers, reused across row tiles.
// A fragments come from an LDS-staged X block (row pad +4 => conflict-free
// bank-pair mapping for the b64 reads under wave32).
// ---------------------------------------------------------------------------
template <int K>
__global__ __launch_bounds__(256) void gcn_gemm_wmma(const float* __restrict__ X,
                                                     const float* __restrict__ W,
                                                     float* __restrict__ Y,
                                                     int nrows) {
  constexpr int KC  = K / 4;    // WMMA K-chunks
  constexpr int LDR = K + 4;    // LDS row stride (floats): stride % 64 == 4
  __shared__ float xs[64 * LDR];

  const int row0 = blockIdx.x * 64;
  const int tid  = threadIdx.x;
  const int lane = tid & 31;
  const int wave = tid >> 5;
  const int hi   = lane >> 4;   // 0: K+0/K+1 halves, 1: K+2/K+3 halves
  const int l16  = lane & 15;
  const int col  = wave * 16 + l16;

  // B fragments: W[k][col] pairs per K-chunk (registers, reused 4x).
  v2f bfrag[KC];
#pragma unroll
  for (int c = 0; c < KC; ++c) {
    const int k = 4 * c + 2 * hi;
    bfrag[c].x = W[k * DH + col];
    bfrag[c].y = W[(k + 1) * DH + col];
  }

  // Cooperative float4 load of the 64 x K X block into LDS.
  {
    constexpr int elems = 64 * K;
    for (int e = tid * 4; e < elems; e += 256 * 4) {
      const int r = e / K;
      const int k = e % K;
      const int gr = row0 + r;
      if (gr < nrows) {
        const float4 v = *(const float4*)&X[(size_t)gr * K + k];
        float* p = &xs[r * LDR + k];
        p[0] = v.x; p[1] = v.y; p[2] = v.z; p[3] = v.w;
      }
    }
  }
  __syncthreads();

#pragma unroll
  for (int rt = 0; rt < 4; ++rt) {
    const int rbase = row0 + rt * 16;
    if (rbase >= nrows) continue;
    v8f acc = {};
    const float* xrow = &xs[(rt * 16 + l16) * LDR + 2 * hi];
#pragma unroll
    for (int c = 0; c < KC; ++c) {
      v2f a;
      a.x = xrow[4 * c + 0];
      a.y = xrow[4 * c + 1];
      // (neg_a, A, neg_b, B, c_mod, C, reuse_a, reuse_b)
      acc = __builtin_amdgcn_wmma_f32_16x16x4_f32(false, a, false, bfrag[c],
                                                  (short)0, acc, false, false);
    }
    // C/D layout: VGPR j -> M = j + 8*hi, N = l16
    float* yout = &Y[(size_t)(rbase + hi * 8) * DH + col];
    if (rbase + 16 <= nrows) {
      // Fast path: whole 16-row tile in range (always taken when nrows%16==0).
#pragma unroll
      for (int j = 0; j < 8; ++j) yout[(size_t)j * DH] = acc[j];
    } else {
#pragma unroll
      for (int j = 0; j < 8; ++j)
        if (rbase + hi * 8 + j < nrows) yout[(size_t)j * DH] = acc[j];
    }
  }
}

// ---------------------------------------------------------------------------
// Degree / norm
// ---------------------------------------------------------------------------
__global__ __launch_bounds__(256) void k_init_deg(float* deg, int n) {
  int i = blockIdx.x * blockDim.x + threadIdx.x;
  if (i < n) deg[i] = 1.0f;  // self-loop contribution
}

__global__ __launch_bounds__(256) void k_deg_scatter(const int* __restrict__ dst,
                                                     float* deg, int E) {
  int e = blockIdx.x * blockDim.x + threadIdx.x;
  if (e < E) atomicAdd(&deg[dst[e]], 1.0f);
}

__global__ __launch_bounds__(256) void k_deg_to_dis(float* deg, int n) {
  int i = blockIdx.x * blockDim.x + threadIdx.x;
  if (i < n) deg[i] = rsqrtf(deg[i]);  // deg >= 1 always (self-loops)
}

// agg[i,:] = t[i,:]*dis[i]^2 + bias[:]  (self-loop term; initializes buffer)
__global__ __launch_bounds__(256) void k_agg_init(const float* __restrict__ t,
                                                  const float* __restrict__ dis,
                                                  const float* __restrict__ bias,
                                                  float* __restrict__ agg, int n) {
  int idx = blockIdx.x * blockDim.x + threadIdx.x;
  int node = idx >> 5;
  int f = (idx & 31) * 4;
  if (node >= n) return;
  const float s2 = dis[node] * dis[node];
  const float4 v = *(const float4*)&t[(size_t)node * DH + f];
  const float4 b = *(const float4*)&bias[f];
  float4 o;
  o.x = v.x * s2 + b.x; o.y = v.y * s2 + b.y;
  o.z = v.z * s2 + b.z; o.w = v.w * s2 + b.w;
  *(float4*)&agg[(size_t)node * DH + f] = o;
}

// One wave per edge, one float4 per lane, 4 f32 atomics per lane.
__global__ __launch_bounds__(256) void k_edge_scatter(const int* __restrict__ src,
                                                      const int* __restrict__ dst,
                                                      const float* __restrict__ dis,
                                                      const float* __restrict__ t,
                                                      float* __restrict__ agg, int E) {
  int gid = blockIdx.x * blockDim.x + threadIdx.x;
  int e = gid >> 5;
  int lane = threadIdx.x & 31;
  if (e >= E) return;
  const int s = src[e];
  const int d = dst[e];
  const float nrm = dis[s] * dis[d];
  const float4 v = *(const float4*)&t[(size_t)s * DH + lane * 4];
  float* ap = &agg[(size_t)d * DH + lane * 4];
  atomicAdd(ap + 0, v.x * nrm);
  atomicAdd(ap + 1, v.y * nrm);
  atomicAdd(ap + 2, v.z * nrm);
  atomicAdd(ap + 3, v.w * nrm);
}

// out = relu((a - m)*rsqrt(v+eps)*g + b) [+ res]; in-place safe (elementwise)
__global__ __launch_bounds__(256) void k_bn_relu(const float* __restrict__ a,
                                                 const float* __restrict__ g,
                                                 const float* __restrict__ bta,
                                                 const float* __restrict__ m,
                                                 const float* __restrict__ var,
                                                 const float* __restrict__ res,
                                                 float* __restrict__ out, int n) {
  int idx = blockIdx.x * blockDim.x + threadIdx.x;
  int node = idx >> 5;
  int f = (idx & 31) * 4;
  if (node >= n) return;
  const float4 x = *(const float4*)&a[(size_t)node * DH + f];
  float4 o;
  {
    float xa[4] = {x.x, x.y, x.z, x.w};
    float oa[4];
#pragma unroll
    for (int c = 0; c < 4; ++c) {
      float y = (xa[c] - m[f + c]) * rsqrtf(var[f + c] + EPSBN) * g[f + c] + bta[f + c];
      oa[c] = fmaxf(y, 0.0f);
    }
    o.x = oa[0]; o.y = oa[1]; o.z = oa[2]; o.w = oa[3];
  }
  if (res) {
    const float4 r = *(const float4*)&res[(size_t)node * DH + f];
    o.x += r.x; o.y += r.y; o.z += r.z; o.w += r.w;
  }
  *(float4*)&out[(size_t)node * DH + f] = o;
}

// ---------------------------------------------------------------------------
// Pooling + MLP head
// ---------------------------------------------------------------------------
__global__ __launch_bounds__(256) void k_pool_zero(float* pooled, float* cnt, int G) {
  int i = blockIdx.x * blockDim.x + threadIdx.x;
  if (i < G * DH) pooled[i] = 0.0f;
  if (i < G) cnt[i] = 0.0f;
}

__global__ __launch_bounds__(256) void k_pool_scatter(const int* __restrict__ batch,
                                                      const float* __restrict__ h,
                                                      float* __restrict__ pooled,
                                                      float* __restrict__ cnt, int n) {
  int idx = blockIdx.x * blockDim.x + threadIdx.x;
  int node = idx >> 5;
  int lane = threadIdx.x & 31;
  if (node >= n) return;
  const int g = batch[node];
  const int f = lane * 4;
  const float4 v = *(const float4*)&h[(size_t)node * DH + f];
  float* pp = &pooled[(size_t)g * DH + f];
  atomicAdd(pp + 0, v.x);
  atomicAdd(pp + 1, v.y);
  atomicAdd(pp + 2, v.z);
  atomicAdd(pp + 3, v.w);
  if (lane == 0) atomicAdd(&cnt[g], 1.0f);
}

__global__ __launch_bounds__(256) void k_pool_div(float* pooled, const float* cnt, int G) {
  int i = blockIdx.x * blockDim.x + threadIdx.x;
  if (i < G * DH) pooled[i] /= fmaxf(cnt[i / DH], 1.0f);
}

// fc1: [G x DH] @ [DH x DH/2] + b, relu
__global__ __launch_bounds__(256) void k_head1(const float* __restrict__ pooled,
                                               const float* __restrict__ w,
                                               const float* __restrict__ b,
                                               float* __restrict__ o, int G) {
  int idx = blockIdx.x * blockDim.x + threadIdx.x;
  if (idx >= G * (DH / 2)) return;
  const int g = idx / (DH / 2);
  const int j = idx % (DH / 2);
  float s = b[j];
#pragma unroll 4
  for (int f = 0; f < DH; ++f) s += pooled[(size_t)g * DH + f] * w[f * (DH / 2) + j];
  o[idx] = fmaxf(s, 0.0f);
}

// fc2: [G x DH/2] @ [DH/2 x 1] + b, sigmoid
__global__ __launch_bounds__(256) void k_head2(const float* __restrict__ h,
                                               const float* __restrict__ w,
                                               const float* __restrict__ b,
                                               float* __restrict__ out, int G) {
  int g = blockIdx.x * blockDim.x + threadIdx.x;
  if (g >= G) return;
  float s = b[0];
#pragma unroll 4
  for (int j = 0; j < DH / 2; ++j) s += h[(size_t)g * (DH / 2) + j] * w[j];
  out[g] = 1.0f / (1.0f + expf(-s));
}

// ---------------------------------------------------------------------------
extern "C" void kernel_launch(void* const* d_in, const int* in_sizes, int n_in,
                              void* d_out, int out_size, void* d_ws, size_t ws_size,
                              hipStream_t stream) {
  const float* x     = (const float*)d_in[0];
  const int*   ei    = (const int*)d_in[1];
  const int*   batch = (const int*)d_in[2];
  const float* W1 = (const float*)d_in[3];  const float* b1 = (const float*)d_in[4];
  const float* g1 = (const float*)d_in[5];  const float* be1 = (const float*)d_in[6];
  const float* m1 = (const float*)d_in[7];  const float* v1 = (const float*)d_in[8];
  const float* W2 = (const float*)d_in[9];  const float* b2 = (const float*)d_in[10];
  const float* g2 = (const float*)d_in[11]; const float* be2 = (const float*)d_in[12];
  const float* m2 = (const float*)d_in[13]; const float* v2 = (const float*)d_in[14];
  const float* W3 = (const float*)d_in[15]; const float* b3 = (const float*)d_in[16];
  const float* g3 = (const float*)d_in[17]; const float* be3 = (const float*)d_in[18];
  const float* m3 = (const float*)d_in[19]; const float* v3 = (const float*)d_in[20];
  const float* f1w = (const float*)d_in[21]; const float* f1b = (const float*)d_in[22];
  const float* f2w = (const float*)d_in[23]; const float* f2b = (const float*)d_in[24];

  const int N = in_sizes[0] / DIN;
  const int E = in_sizes[1] / 2;
  const int G = out_size;
  const int* src = ei;
  const int* dst = ei + E;

  float* ws = (float*)d_ws;
  size_t off = 0;
  float* dis    = ws + off; off += (size_t)N;
  float* bufA   = ws + off; off += (size_t)N * DH;
  float* bufB   = ws + off; off += (size_t)N * DH;
  float* bufC   = ws + off; off += (size_t)N * DH;
  float* pooled = ws + off; off += (size_t)G * DH;
  float* cnt    = ws + off; off += (size_t)G;
  float* fc1o   = ws + off; off += (size_t)G * (DH / 2);

  const int TB = 256;
  const int gN   = (N + TB - 1) / TB;            // per-node
  const int gE   = (E + TB - 1) / TB;            // per-edge
  const int gNw  = (N * 32 + TB - 1) / TB;       // wave-per-node
  const int gEw  = (E * 32 + TB - 1) / TB;       // wave-per-edge
  const int gGem = (N + 63) / 64;                // GEMM row blocks

  // degree -> dis (self-loops included via init=1)
  k_init_deg<<<gN, TB, 0, stream>>>(dis, N);
  k_deg_scatter<<<gE, TB, 0, stream>>>(dst, dis, E);
  k_deg_to_dis<<<gN, TB, 0, stream>>>(dis, N);

  // ----- layer 1: t1 = x@W1 -> aggregate -> bn+relu (h1 in bufB) -----
  gcn_gemm_wmma<DIN><<<gGem, TB, 0, stream>>>(x, W1, bufA, N);
  k_agg_init<<<gNw, TB, 0, stream>>>(bufA, dis, b1, bufB, N);
  k_edge_scatter<<<gEw, TB, 0, stream>>>(src, dst, dis, bufA, bufB, E);
  k_bn_relu<<<gNw, TB, 0, stream>>>(bufB, g1, be1, m1, v1, nullptr, bufB, N);

  // ----- layer 2 (residual): t2 = h1@W2 -> agg -> bn+relu + h1 (h2 in bufC) --
  gcn_gemm_wmma<DH><<<gGem, TB, 0, stream>>>(bufB, W2, bufA, N);
  k_agg_init<<<gNw, TB, 0, stream>>>(bufA, dis, b2, bufC, N);
  k_edge_scatter<<<gEw, TB, 0, stream>>>(src, dst, dis, bufA, bufC, E);
  k_bn_relu<<<gNw, TB, 0, stream>>>(bufC, g2, be2, m2, v2, bufB, bufC, N);

  // ----- layer 3: t3 = h2@W3 -> agg -> bn+relu (h3 in bufB) -----
  gcn_gemm_wmma<DH><<<gGem, TB, 0, stream>>>(bufC, W3, bufA, N);
  k_agg_init<<<gNw, TB, 0, stream>>>(bufA, dis, b3, bufB, N);
  k_edge_scatter<<<gEw, TB, 0, stream>>>(src, dst, dis, bufA, bufB, E);
  k_bn_relu<<<gNw, TB, 0, stream>>>(bufB, g3, be3, m3, v3, nullptr, bufB, N);

  // ----- global mean pool + MLP head -----
  k_pool_zero<<<(G * DH + TB - 1) / TB, TB, 0, stream>>>(pooled, cnt, G);
  k_pool_scatter<<<gNw, TB, 0, stream>>>(batch, bufB, pooled, cnt, N);
  k_pool_div<<<(G * DH + TB - 1) / TB, TB, 0, stream>>>(pooled, cnt, G);
  k_head1<<<(G * (DH / 2) + TB - 1) / TB, TB, 0, stream>>>(pooled, f1w, f1b, fc1o, G);
  k_head2<<<(G + TB - 1) / TB, TB, 0, stream>>>(fc1o, f2w, f2b, (float*)d_out, G);
}